// WassersteinExp_67525475827936
// MI455X (gfx1250) — compile-verified
//
#include <hip/hip_runtime.h>
#include <math.h>

#define TPB 256
#define WEPS 1e-8f

#if __has_builtin(__builtin_amdgcn_global_load_async_to_lds_b32)
#define HAVE_ASYNC_COPY 1
typedef __attribute__((address_space(1))) int* gas1_t;   // global (device) int*
typedef __attribute__((address_space(3))) int* las3_t;   // LDS (shared) int*
#endif

__device__ __forceinline__ float frcp(float x) { return __builtin_amdgcn_rcpf(x); }
__device__ __forceinline__ float frsq(float x) { return __builtin_amdgcn_rsqf(x); }

__global__ void __launch_bounds__(TPB)
wasserstein_exp_kernel(const float* __restrict__ loc,
                       const float* __restrict__ cov1,
                       const float* __restrict__ vel,
                       const float* __restrict__ vcov,
                       float* __restrict__ out_loc,
                       float* __restrict__ out_cov,
                       int Bn)
{
    __shared__ float s_c1[TPB * 9];
    __shared__ float s_vc[TPB * 9];
    __shared__ float s_l[TPB * 3];
    __shared__ float s_v[TPB * 3];

    const int t = threadIdx.x;
    const int base = blockIdx.x * TPB;
    const int elem = base + t;
    const int max9 = Bn * 9 - 1;
    const int max3 = Bn * 3 - 1;

    // ---- Stage AoS inputs into LDS with coalesced async copies (CDNA5 path) ----
#ifdef HAVE_ASYNC_COPY
#pragma unroll
    for (int j = 0; j < 9; ++j) {
        int g = base * 9 + j * TPB + t; g = g > max9 ? max9 : g;
        __builtin_amdgcn_global_load_async_to_lds_b32((gas1_t)(cov1 + g),
                                                      (las3_t)(&s_c1[j * TPB + t]), 0, 0);
        __builtin_amdgcn_global_load_async_to_lds_b32((gas1_t)(vcov + g),
                                                      (las3_t)(&s_vc[j * TPB + t]), 0, 0);
    }
#pragma unroll
    for (int j = 0; j < 3; ++j) {
        int g = base * 3 + j * TPB + t; g = g > max3 ? max3 : g;
        __builtin_amdgcn_global_load_async_to_lds_b32((gas1_t)(loc + g),
                                                      (las3_t)(&s_l[j * TPB + t]), 0, 0);
        __builtin_amdgcn_global_load_async_to_lds_b32((gas1_t)(vel + g),
                                                      (las3_t)(&s_v[j * TPB + t]), 0, 0);
    }
#if __has_builtin(__builtin_amdgcn_s_wait_asynccnt)
    __builtin_amdgcn_s_wait_asynccnt(0);
#else
    asm volatile("s_wait_asynccnt 0" ::: "memory");
#endif
#else
    // Fallback: plain coalesced loads staged through LDS
#pragma unroll
    for (int j = 0; j < 9; ++j) {
        int g = base * 9 + j * TPB + t; g = g > max9 ? max9 : g;
        s_c1[j * TPB + t] = cov1[g];
        s_vc[j * TPB + t] = vcov[g];
    }
#pragma unroll
    for (int j = 0; j < 3; ++j) {
        int g = base * 3 + j * TPB + t; g = g > max3 ? max3 : g;
        s_l[j * TPB + t] = loc[g];
        s_v[j * TPB + t] = vel[g];
    }
#endif
    __syncthreads();

    // ---- Load this element's record from LDS (stride-9: conflict-free) ----
    const int o9 = t * 9, o3 = t * 3;
    const float a00 = s_c1[o9 + 0], a01 = s_c1[o9 + 1], a02 = s_c1[o9 + 2];
    const float a11 = s_c1[o9 + 4], a12 = s_c1[o9 + 5], a22 = s_c1[o9 + 8];
    const float v00 = s_vc[o9 + 0], v01 = s_vc[o9 + 1], v02 = s_vc[o9 + 2];
    const float v11 = s_vc[o9 + 4], v12 = s_vc[o9 + 5], v22 = s_vc[o9 + 8];
    const float lx = s_l[o3 + 0], ly = s_l[o3 + 1], lz = s_l[o3 + 2];
    const float vlx = s_v[o3 + 0], vly = s_v[o3 + 1], vlz = s_v[o3 + 2];

    // ================= Analytic symmetric 3x3 eigendecomposition =================
    const float q  = (a00 + a11 + a22) * (1.0f / 3.0f);
    const float p1 = a01 * a01 + a02 * a02 + a12 * a12;
    const float d0 = a00 - q, d1 = a11 - q, d2 = a22 - q;
    const float p2 = d0 * d0 + d1 * d1 + d2 * d2 + 2.0f * p1;
    const float pp = sqrtf(p2 * (1.0f / 6.0f));
    const bool  iso = p2 < 1e-24f;
    const float invp = iso ? 0.0f : frcp(pp);

    const float b00 = d0 * invp, b01 = a01 * invp, b02 = a02 * invp;
    const float b11 = d1 * invp, b12 = a12 * invp, b22 = d2 * invp;
    float detB = b00 * (b11 * b22 - b12 * b12)
               - b01 * (b01 * b22 - b12 * b02)
               + b02 * (b01 * b12 - b11 * b02);
    float r = 0.5f * detB;
    r = fminf(1.0f, fmaxf(-1.0f, r));
    const float phi = acosf(r) * (1.0f / 3.0f);
    const float c0 = __cosf(phi);
    const float c2 = __cosf(phi + 2.0943951023931953f); // + 2*pi/3
    const float e0 = q + 2.0f * pp * c0;                // largest
    const float e2 = q + 2.0f * pp * c2;                // smallest
    const float e1 = 3.0f * q - e0 - e2;                // middle

    // Primary eigenvector: pick the more isolated extreme eigenvalue (rank-2 M)
    const float la = ((e0 - e1) >= (e1 - e2)) ? e0 : e2;
    const float m00 = a00 - la, m11 = a11 - la, m22 = a22 - la;
    // cross products of rows of (A - la*I)
    const float cx0 = a01 * a12 - a02 * m11, cy0 = a02 * a01 - m00 * a12, cz0 = m00 * m11 - a01 * a01;
    const float cx1 = a01 * m22 - a02 * a12, cy1 = a02 * a02 - m00 * m22, cz1 = m00 * a12 - a01 * a02;
    const float cx2 = m11 * m22 - a12 * a12, cy2 = a12 * a02 - a01 * m22, cz2 = a01 * a12 - m11 * a02;
    const float n0 = cx0 * cx0 + cy0 * cy0 + cz0 * cz0;
    const float n1 = cx1 * cx1 + cy1 * cy1 + cz1 * cz1;
    const float n2 = cx2 * cx2 + cy2 * cy2 + cz2 * cz2;
    float bx = cx0, by = cy0, bz = cz0, bn = n0;
    if (n1 > bn) { bx = cx1; by = cy1; bz = cz1; bn = n1; }
    if (n2 > bn) { bx = cx2; by = cy2; bz = cz2; bn = n2; }
    const bool degen = iso || (bn < 1e-28f);
    const float inl = degen ? 0.0f : frsq(bn);
    const float vax = degen ? 1.0f : bx * inl;
    const float vay = degen ? 0.0f : by * inl;
    const float vaz = degen ? 0.0f : bz * inl;

    // Orthonormal completion {u, w0} of span{va}^perp
    float ux, uy, uz;
    if (fabsf(vax) > fabsf(vaz)) {
        const float si = frsq(vax * vax + vay * vay);
        ux = -vay * si; uy = vax * si; uz = 0.0f;
    } else {
        const float si = frsq(vay * vay + vaz * vaz);
        ux = 0.0f; uy = -vaz * si; uz = vay * si;
    }
    const float w0x = vay * uz - vaz * uy;
    const float w0y = vaz * ux - vax * uz;
    const float w0z = vax * uy - vay * ux;

    // Exact 2x2 eigenproblem of A restricted to span{u, w0}
    const float Aux = a00 * ux + a01 * uy + a02 * uz;
    const float Auy = a01 * ux + a11 * uy + a12 * uz;
    const float Auz = a02 * ux + a12 * uy + a22 * uz;
    const float Awx = a00 * w0x + a01 * w0y + a02 * w0z;
    const float Awy = a01 * w0x + a11 * w0y + a12 * w0z;
    const float Awz = a02 * w0x + a12 * w0y + a22 * w0z;
    const float auu = ux * Aux + uy * Auy + uz * Auz;
    const float auw = w0x * Aux + w0y * Auy + w0z * Auz;
    const float aww = w0x * Awx + w0y * Awy + w0z * Awz;
    const float hh = 0.5f * (auu - aww);
    const float rt = sqrtf(hh * hh + auw * auw);
    const float mid = 0.5f * (auu + aww);
    const float lb = mid + rt;
    const float lc = mid - rt;
    // eigenvector of [[auu,auw],[auw,aww]] for lb: (auw, lb-auu) or (lb-aww, auw)
    const float p0x = auw,      p0y = lb - auu;
    const float q0x = lb - aww, q0y = auw;
    const float np = p0x * p0x + p0y * p0y;
    const float nq = q0x * q0x + q0y * q0y;
    float cc, ssn;
    if (nq > np)            { const float is_ = frsq(nq); cc = q0x * is_; ssn = q0y * is_; }
    else if (np > 1e-30f)   { const float is_ = frsq(np); cc = p0x * is_; ssn = p0y * is_; }
    else                    { cc = 1.0f; ssn = 0.0f; }

    // R columns = eigenvectors; s = matching eigenvalues
    float R[3][3], s[3];
    R[0][0] = vax; R[1][0] = vay; R[2][0] = vaz;
    R[0][1] =  cc * ux + ssn * w0x; R[1][1] =  cc * uy + ssn * w0y; R[2][1] =  cc * uz + ssn * w0z;
    R[0][2] = -ssn * ux + cc * w0x; R[1][2] = -ssn * uy + cc * w0y; R[2][2] = -ssn * uz + cc * w0z;
    s[0] = la; s[1] = lb; s[2] = lc;

    // ====================== Wasserstein exponential map ======================
    const float V[3][3] = {{v00, v01, v02}, {v01, v11, v12}, {v02, v12, v22}};

    // T = V * R ; C = R^T * T
    float T[3][3], C[3][3];
#pragma unroll
    for (int i = 0; i < 3; ++i)
#pragma unroll
        for (int j = 0; j < 3; ++j)
            T[i][j] = V[i][0] * R[0][j] + V[i][1] * R[1][j] + V[i][2] * R[2][j];
#pragma unroll
    for (int i = 0; i < 3; ++i)
#pragma unroll
        for (int j = 0; j < 3; ++j)
            C[i][j] = R[0][i] * T[0][j] + R[1][i] * T[1][j] + R[2][i] * T[2][j];

    // E = C / (s_i + s_j + eps)   (6 unique reciprocals)
    float DI[3][3];
#pragma unroll
    for (int i = 0; i < 3; ++i)
#pragma unroll
        for (int j = i; j < 3; ++j) {
            const float rr = frcp(s[i] + s[j] + WEPS);
            DI[i][j] = rr; DI[j][i] = rr;
        }
    float E[3][3];
#pragma unroll
    for (int i = 0; i < 3; ++i)
#pragma unroll
        for (int j = 0; j < 3; ++j)
            E[i][j] = C[i][j] * DI[i][j];

    // gamma = R * E * R^T
    float G[3][3], gam[3][3];
#pragma unroll
    for (int i = 0; i < 3; ++i)
#pragma unroll
        for (int j = 0; j < 3; ++j)
            G[i][j] = R[i][0] * E[0][j] + R[i][1] * E[1][j] + R[i][2] * E[2][j];
#pragma unroll
    for (int i = 0; i < 3; ++i)
#pragma unroll
        for (int j = 0; j < 3; ++j)
            gam[i][j] = G[i][0] * R[j][0] + G[i][1] * R[j][1] + G[i][2] * R[j][2];

    // cov = R * diag(s) * R^T
    float RS[3][3], cov[3][3];
#pragma unroll
    for (int i = 0; i < 3; ++i)
#pragma unroll
        for (int k = 0; k < 3; ++k)
            RS[i][k] = R[i][k] * s[k];
#pragma unroll
    for (int i = 0; i < 3; ++i)
#pragma unroll
        for (int j = 0; j < 3; ++j)
            cov[i][j] = RS[i][0] * R[j][0] + RS[i][1] * R[j][1] + RS[i][2] * R[j][2];

    // P = gamma * cov * gamma^T
    float H[3][3], P[3][3];
#pragma unroll
    for (int i = 0; i < 3; ++i)
#pragma unroll
        for (int j = 0; j < 3; ++j)
            H[i][j] = gam[i][0] * cov[0][j] + gam[i][1] * cov[1][j] + gam[i][2] * cov[2][j];
#pragma unroll
    for (int i = 0; i < 3; ++i)
#pragma unroll
        for (int j = 0; j < 3; ++j)
            P[i][j] = H[i][0] * gam[j][0] + H[i][1] * gam[j][1] + H[i][2] * gam[j][2];

    // ---- Store results ----
    if (elem < Bn) {
        float* ol = out_loc + elem * 3;
        ol[0] = lx + vlx; ol[1] = ly + vly; ol[2] = lz + vlz;
        float* oc = out_cov + elem * 9;
#pragma unroll
        for (int i = 0; i < 3; ++i)
#pragma unroll
            for (int j = 0; j < 3; ++j)
                oc[i * 3 + j] = cov[i][j] + V[i][j] + P[i][j];
    }
}

extern "C" void kernel_launch(void* const* d_in, const int* in_sizes, int n_in,
                              void* d_out, int out_size, void* d_ws, size_t ws_size,
                              hipStream_t stream) {
    const float* loc  = (const float*)d_in[0];  // [B,3]
    const float* cov1 = (const float*)d_in[1];  // [B,3,3]
    const float* vel  = (const float*)d_in[2];  // [B,3]
    const float* vcov = (const float*)d_in[3];  // [B,3,3]
    const int Bn = in_sizes[0] / 3;

    float* out     = (float*)d_out;
    float* out_loc = out;                        // B*3 floats
    float* out_cov = out + (size_t)Bn * 3;       // B*9 floats

    const int blocks = (Bn + TPB - 1) / TPB;
    wasserstein_exp_kernel<<<blocks, TPB, 0, stream>>>(loc, cov1, vel, vcov,
                                                       out_loc, out_cov, Bn);
}